// BoothLinear_27625229648225
// MI455X (gfx1250) — compile-verified
//
#include <hip/hip_runtime.h>
#include <stdint.h>

// ---------------------------------------------------------------------------
// Booth elementwise kernel for MI455X (gfx1250, wave32).
//
// Roofline: N = 33.55M elems, 12 B/elem = 402 MB -> ~17.3 us @ 23.3 TB/s.
// A scalar int32 Booth loop is ~95 VALU/elem (~2.3x over the memory roof), so
// two elements are packed per 32-bit register and the loop runs on VOP3P
// packed 16-bit math (v_pk_sub_i16 / v_pk_mul_lo_u16 / v_pk_ashrrev_i16 /
// v_pk_lshrrev_b16) -> ~55 VALU/elem, back at the bandwidth roof.
//
// No WMMA: this op is elementwise (no contraction dim). The CDNA5 data-
// movement feature exercised is the async global->LDS engine:
//   GLOBAL_LOAD_ASYNC_TO_LDS_B128 (inline asm; the clang builtin's signature
//   is toolchain-dependent) + ASYNCcnt + s_wait_asynccnt,
// double-buffered per 2048-element block tile (32 KB LDS of 320 KB/WGP),
// overlapping next-tile staging with current-tile packed compute.
// Output uses b128 nontemporal stores (pure stream, keep 192MB L2 clean).
// ---------------------------------------------------------------------------

#define AS1 __attribute__((address_space(1)))
#define AS3 __attribute__((address_space(3)))

#if defined(__AMDGCN__) && defined(__gfx1250__)
#define HAVE_ASYNC_LDS 1
#else
#define HAVE_ASYNC_LDS 0
#endif

typedef short          s16x2 __attribute__((ext_vector_type(2)));
typedef unsigned short u16x2 __attribute__((ext_vector_type(2)));
typedef float          f32x4 __attribute__((ext_vector_type(4)));

struct FPair { float a, b; };

static __device__ __forceinline__ uint32_t pk_sub_i16(uint32_t a, uint32_t b) {
    s16x2 r = __builtin_bit_cast(s16x2, a) - __builtin_bit_cast(s16x2, b);
    return __builtin_bit_cast(uint32_t, r);
}
static __device__ __forceinline__ uint32_t pk_mul_lo_u16(uint32_t a, uint32_t b) {
    u16x2 r = __builtin_bit_cast(u16x2, a) * __builtin_bit_cast(u16x2, b);
    return __builtin_bit_cast(uint32_t, r);
}
static __device__ __forceinline__ uint32_t pk_ashr1_i16(uint32_t a) {
    s16x2 r = __builtin_bit_cast(s16x2, a) >> 1;
    return __builtin_bit_cast(uint32_t, r);
}
static __device__ __forceinline__ uint32_t pk_lshr1_u16(uint32_t a) {
    u16x2 r = __builtin_bit_cast(u16x2, a) >> 1;
    return __builtin_bit_cast(uint32_t, r);
}

// Booth forward (bits == 8) on two packed elements. Per 16-bit lane this
// exactly mirrors the reference int32 recurrence:
//   lb = q & 1
//   a  = (a - (lb - q_prev) * m) & 0xFF      (sub on 0->1, add on 1->0)
//   q  = (q >>a 1) | ((a & 1) << 7)
//   a  = (a >> 1) | (a & 0x80)
// Returns packed per-lane r16 = (a << 8) | q16.
static __device__ __forceinline__ uint32_t booth8_pair(uint32_t m, uint32_t q) {
    uint32_t a = 0u, p = 0u;
#pragma unroll
    for (int i = 0; i < 8; ++i) {
        const uint32_t lb = q & 0x00010001u;                 // last bit per lane
        const uint32_t d  = pk_sub_i16(lb, p);               // {-1,0,+1} per lane
        const uint32_t t  = pk_mul_lo_u16(d, m);             // d*m (mod 2^16)
        const uint32_t a2 = pk_sub_i16(a, t) & 0x00FF00FFu;  // (a - d*m) & 0xFF
        p = lb;
        // bit0/bit16 shifted to bit7/bit23: no cross-lane contamination.
        q = pk_ashr1_i16(q) | ((a2 & 0x00010001u) << 7);
        a = pk_lshr1_u16(a2) | (a2 & 0x00800080u);
    }
    // a lanes <= 0xFF, so a<<8 stays inside each 16-bit lane.
    return (a << 8) | q;
}

// result_ref = r16(zext) | (q0 >> 8)  -- exact for |q0| < 32768
// (negative weights: q0>>8 == -1 -> result == -1 -> -1.0f - 65536 = -65537).
static __device__ __forceinline__ float booth_fin(uint32_t r16, int32_t q_shr8) {
    const int32_t result = (int32_t)r16 | q_shr8;
    const float f = (float)result;
    return (result & 0x8000) ? (f - 65536.0f) : f;
}

// Exact dynamic-`bits` scalar fallback (full int32 semantics).
static __device__ float booth_scalar(float xf, float wf, int bits) {
    int32_t m = (int32_t)rintf(xf);
    int32_t q = (int32_t)rintf(wf);
    const uint32_t mask = (bits >= 32) ? 0xFFFFFFFFu : ((1u << bits) - 1u);
    int32_t a = 0;
    int32_t qp = 0;
    for (int k = 0; k < bits; ++k) {
        const int32_t lb = q & 1;
        if (lb == 1 && qp == 0)      a = (int32_t)(((uint32_t)(a - m)) & mask);
        else if (lb == 0 && qp == 1) a = (int32_t)(((uint32_t)(a + m)) & mask);
        qp = lb;
        q = (q >> 1) | ((a & 1) << (bits - 1));
        a = (a >> 1) | (int32_t)((uint32_t)a & (1u << (bits - 1)));
    }
    const int32_t result = (int32_t)(((uint32_t)a << bits) | (uint32_t)q);
    float f = (float)result;
    if ((uint32_t)result & (1u << (2 * bits - 1))) f -= (float)(1ll << (2 * bits));
    return f;
}

static __device__ __forceinline__ FPair booth8_two(float xa, float xb, float wa, float wb) {
    const int32_t mi0 = (int32_t)rintf(xa);
    const int32_t mi1 = (int32_t)rintf(xb);
    const int32_t qi0 = (int32_t)rintf(wa);
    const int32_t qi1 = (int32_t)rintf(wb);
    const uint32_t m  = ((uint32_t)mi0 & 0xFFFFu) | ((uint32_t)mi1 << 16);
    const uint32_t qq = ((uint32_t)qi0 & 0xFFFFu) | ((uint32_t)qi1 << 16);
    const uint32_t r  = booth8_pair(m, qq);
    FPair o;
    o.a = booth_fin(r & 0xFFFFu, ((int32_t)(qq << 16)) >> 24);  // sext(q0) >> 8
    o.b = booth_fin(r >> 16,     ((int32_t)qq) >> 24);          // sext(q1) >> 8
    return o;
}

// Compute 8 elements (two f32x4 of x and w) and NT-store them.
static __device__ __forceinline__ void compute8_store(const f32x4 xv0, const f32x4 xv1,
                                                      const f32x4 wv0, const f32x4 wv1,
                                                      float* __restrict__ outp) {
    const FPair p0 = booth8_two(xv0[0], xv0[1], wv0[0], wv0[1]);
    const FPair p1 = booth8_two(xv0[2], xv0[3], wv0[2], wv0[3]);
    const FPair p2 = booth8_two(xv1[0], xv1[1], wv1[0], wv1[1]);
    const FPair p3 = booth8_two(xv1[2], xv1[3], wv1[2], wv1[3]);
    f32x4 o0, o1;
    o0[0] = p0.a; o0[1] = p0.b; o0[2] = p1.a; o0[3] = p1.b;
    o1[0] = p2.a; o1[1] = p2.b; o1[2] = p3.a; o1[3] = p3.b;
    __builtin_nontemporal_store(o0, (f32x4*)outp);
    __builtin_nontemporal_store(o1, (f32x4*)(outp + 4));
}

// One 8-element chunk straight from global memory (fast or scalar tail).
static __device__ __forceinline__ void chunk_direct(const float* __restrict__ x,
                                                    const float* __restrict__ w,
                                                    float* __restrict__ out,
                                                    int base, int n, int bits) {
    if (bits == 8 && base + 8 <= n) {
        const f32x4 xv0 = __builtin_nontemporal_load((const f32x4*)(x + base));
        const f32x4 xv1 = __builtin_nontemporal_load((const f32x4*)(x + base + 4));
        const f32x4 wv0 = __builtin_nontemporal_load((const f32x4*)(w + base));
        const f32x4 wv1 = __builtin_nontemporal_load((const f32x4*)(w + base + 4));
        compute8_store(xv0, xv1, wv0, wv1, out + base);
    } else {
        const int rem = n - base;
        const int lim = rem < 8 ? rem : 8;
        for (int j = 0; j < lim; ++j)
            out[base + j] = booth_scalar(x[base + j], w[base + j], bits);
    }
}

#if HAVE_ASYNC_LDS
// Async copy 16B/lane global -> LDS, tracked by ASYNCcnt. The clang builtin's
// prototype differs across toolchains (probe: param0 is v4i on this one), so
// use the ISA mnemonic directly: VDST = LDS byte address, VADDR = 64-bit
// global address ("GV" mode, saddr=off).
static __device__ __forceinline__ void async_ld_b128(const float* g, float* l) {
    const uint32_t lds_addr = (uint32_t)(uintptr_t)(AS3 void*)l;  // real LDS offset
    asm volatile("global_load_async_to_lds_b128 %0, %1, off"
                 :: "v"(lds_addr), "v"(g)
                 : "memory");
}
template <int N> static __device__ __forceinline__ void wait_async() {
#if __has_builtin(__builtin_amdgcn_s_wait_asynccnt)
    __builtin_amdgcn_s_wait_asynccnt(N);
#else
    asm volatile("s_wait_asynccnt %0" :: "i"(N) : "memory");
#endif
}
#endif

// Tile = 256 threads * 8 elems = 2048 elements (8 KB per array).
__global__ __launch_bounds__(256) void booth_kernel(
    const float* __restrict__ x, const float* __restrict__ w,
    float* __restrict__ out, const int* __restrict__ bits_p, int n)
{
#if HAVE_ASYNC_LDS
    __shared__ __attribute__((aligned(16))) float lx[2][2048];
    __shared__ __attribute__((aligned(16))) float lw[2][2048];

    const int bits    = *bits_p;               // uniform -> scalar load
    const int nChunks = (n + 7) >> 3;
    const int nTiles  = (nChunks + 255) >> 8;
    const int tid     = (int)threadIdx.x;

    int tile = (int)blockIdx.x;
    if (tile >= nTiles) return;

    int  buf     = 0;
    bool curFull = (bits == 8) && ((tile + 1) * 2048 <= n);  // uniform
    if (curFull) {
        const int eb = tile * 2048 + tid * 8;
        async_ld_b128(x + eb,     &lx[0][tid * 8]);
        async_ld_b128(x + eb + 4, &lx[0][tid * 8 + 4]);
        async_ld_b128(w + eb,     &lw[0][tid * 8]);
        async_ld_b128(w + eb + 4, &lw[0][tid * 8 + 4]);
    }

    for (; tile < nTiles; tile += (int)gridDim.x) {
        const int  ntile    = tile + (int)gridDim.x;
        const bool nextFull = (ntile < nTiles) && (bits == 8) &&
                              ((ntile + 1) * 2048 <= n);     // uniform
        if (nextFull) {  // stage next tile while computing current
            const int eb = ntile * 2048 + tid * 8;
            const int nb = buf ^ 1;
            async_ld_b128(x + eb,     &lx[nb][tid * 8]);
            async_ld_b128(x + eb + 4, &lx[nb][tid * 8 + 4]);
            async_ld_b128(w + eb,     &lw[nb][tid * 8]);
            async_ld_b128(w + eb + 4, &lw[nb][tid * 8 + 4]);
        }
        if (curFull) {
            // Async loads complete in order: <=4 outstanding => current
            // buffer's 4 loads have landed in LDS.
            if (nextFull) wait_async<4>(); else wait_async<0>();
            const f32x4* lxv = (const f32x4*)&lx[buf][tid * 8];
            const f32x4* lwv = (const f32x4*)&lw[buf][tid * 8];
            const f32x4 xv0 = lxv[0], xv1 = lxv[1];   // ds_load_b128
            const f32x4 wv0 = lwv[0], wv1 = lwv[1];
            compute8_store(xv0, xv1, wv0, wv1, out + tile * 2048 + tid * 8);
        } else {
            chunk_direct(x, w, out, tile * 2048 + tid * 8, n, bits);
        }
        buf ^= 1;
        curFull = nextFull;
    }
#else
    // Fallback (and host-pass parse body): plain grid-stride streaming kernel.
    const int bits    = *bits_p;
    const int nChunks = (n + 7) >> 3;
    const int stride  = (int)(gridDim.x * blockDim.x);
    for (int c = (int)(blockIdx.x * blockDim.x + threadIdx.x); c < nChunks; c += stride) {
        const int base = c << 3;
        const int pf = base + (stride << 3);
        if (pf + 8 <= n) {
            __builtin_prefetch(x + pf, 0, 0);
            __builtin_prefetch(w + pf, 0, 0);
        }
        chunk_direct(x, w, out, base, n, bits);
    }
#endif
}

extern "C" void kernel_launch(void* const* d_in, const int* in_sizes, int n_in,
                              void* d_out, int out_size, void* d_ws, size_t ws_size,
                              hipStream_t stream) {
    (void)n_in; (void)d_ws; (void)ws_size; (void)out_size;
    const float* x      = (const float*)d_in[0];
    const float* w      = (const float*)d_in[1];
    const int*   bits_p = (const int*)d_in[2];
    float*       out    = (float*)d_out;

    const int n = in_sizes[0];                        // 4096*8192 = 33,554,432
    const long long nChunks = ((long long)n + 7) / 8; // 8 elems per thread-chunk
    const long long nTiles  = (nChunks + 255) / 256;  // 2048-elem block tiles
    // Cap at 8192 blocks: 2M threads in flight, and >=2 tiles per block on the
    // full problem so the async double-buffer actually overlaps.
    long long g = nTiles < 8192 ? nTiles : 8192;
    int grid = (int)(g < 1 ? 1 : g);

    booth_kernel<<<grid, 256, 0, stream>>>(x, w, out, bits_p, n);
}